// LorentzAvgPool2d_47742856463156
// MI455X (gfx1250) — compile-verified
//
#include <hip/hip_runtime.h>
#include <cstdint>

// LorentzAvgPool2d: [32,128,128,64] f32 -> [32,64,64,64] f32
// 2x2/stride-2 mean pool + Lorentz centroid normalization (k=1).
// Pure HBM-streaming kernel; data moved via gfx1250 async global->LDS copies.

typedef float f4 __attribute__((ext_vector_type(4)));

#define POOL_B   32
#define POOL_H   128
#define POOL_W   128
#define POOL_C   64
#define POOL_OH  64
#define POOL_OW  64
#define EPS_V    1e-8f

__global__ __launch_bounds__(256) void lorentz_pool_kernel(
    const float* __restrict__ x, float* __restrict__ out) {
  // 8 waves * 4 staged b128 loads * 32 lanes * 16B = 16 KB
  __shared__ float smem[8 * 512];

  const unsigned tid  = threadIdx.x;
  const unsigned lane = tid & 31u;
  const unsigned wave = tid >> 5;
  const unsigned g    = blockIdx.x * 256u + tid;

  const unsigned chunk = g & 15u;        // which float4 of the 64 channels
  const unsigned pix   = g >> 4;         // output pixel index
  const unsigned ow    = pix & (POOL_OW - 1);
  const unsigned oh    = (pix >> 6) & (POOL_OH - 1);
  const unsigned bb    = pix >> 12;

  const size_t rowstride = (size_t)POOL_W * POOL_C;  // floats per input row
  const size_t base =
      (((size_t)bb * POOL_H + 2 * oh) * POOL_W + 2 * ow) * POOL_C + 4 * chunk;

  const float* p00 = x + base;
  const float* p01 = x + base + POOL_C;
  const float* p10 = x + base + rowstride;
  const float* p11 = x + base + rowstride + POOL_C;

  // Wave-relative LDS byte address of smem (flat LDS aperture is 4GB-aligned;
  // low 32 bits of the generic pointer == wave-relative LDS offset, which is
  // exactly what the async op's VDST address expects).
  const uint32_t ldsbase = (uint32_t)(uintptr_t)&smem[0];
  // Conflict-free staging: slot(wave,k,lane) = wave*2048 + k*512 + lane*16
  const uint32_t myofs = ldsbase + wave * 2048u + lane * 16u;

  asm volatile("global_load_async_to_lds_b128 %0, %1, off"
               :: "v"(myofs +    0u), "v"(p00) : "memory");
  asm volatile("global_load_async_to_lds_b128 %0, %1, off"
               :: "v"(myofs +  512u), "v"(p01) : "memory");
  asm volatile("global_load_async_to_lds_b128 %0, %1, off"
               :: "v"(myofs + 1024u), "v"(p10) : "memory");
  asm volatile("global_load_async_to_lds_b128 %0, %1, off"
               :: "v"(myofs + 1536u), "v"(p11) : "memory");
  asm volatile("s_wait_asynccnt 0x0" ::: "memory");

  // Read back the 4 staged chunks (lane-contiguous -> conflict-free b128 reads;
  // same wave wrote them, so only asynccnt ordering is needed, no barrier).
  const unsigned fbase = wave * 512u + lane * 4u;
  f4 a0 = *(const f4*)&smem[fbase +   0u];
  f4 a1 = *(const f4*)&smem[fbase + 128u];
  f4 a2 = *(const f4*)&smem[fbase + 256u];
  f4 a3 = *(const f4*)&smem[fbase + 384u];

  f4 avg = ((a0 + a1) + (a2 + a3)) * 0.25f;

  // Partial Minkowski norm: channel 0 (time coord) enters with a minus sign.
  float part = avg.x * avg.x + avg.y * avg.y + avg.z * avg.z + avg.w * avg.w;
  if (chunk == 0u) part -= 2.0f * avg.x * avg.x;

  // Reduce across the 16 lanes sharing this pixel (xor masks stay in-group).
  part += __shfl_xor(part, 1);
  part += __shfl_xor(part, 2);
  part += __shfl_xor(part, 4);
  part += __shfl_xor(part, 8);
  // part == inner = -a0^2 + sum_{c>0} a_c^2 ; |-inner| == |inner|
  const float denom = sqrtf(fmaxf(fabsf(part), EPS_V));
  const float scale = 1.0f / denom;  // sqrt(k)=1

  f4 o = avg * scale;
  *(f4*)(out + (size_t)pix * POOL_C + 4u * chunk) = o;
}

extern "C" void kernel_launch(void* const* d_in, const int* in_sizes, int n_in,
                              void* d_out, int out_size, void* d_ws, size_t ws_size,
                              hipStream_t stream) {
  const float* x = (const float*)d_in[0];
  float* out = (float*)d_out;

  // 32*64*64 output pixels * 16 chunks = 2,097,152 threads; exact full blocks.
  const unsigned total = POOL_B * POOL_OH * POOL_OW * 16u;
  const unsigned block = 256u;
  const unsigned grid  = total / block;  // 8192

  lorentz_pool_kernel<<<grid, block, 0, stream>>>(x, out);
}